// ann1_snn2_45535243272632
// MI455X (gfx1250) — compile-verified
//
#include <hip/hip_runtime.h>
#include <math.h>
#include <stdint.h>

typedef __attribute__((ext_vector_type(2))) float v2f;
typedef __attribute__((ext_vector_type(8))) float v8f;

// ---------------------------------------------------------------------------
// GEMM over "bt" rows:  Y[b, h, t] = act( sum_k A[b, k, t] * W[h, k] + bias[h] )
// A: [Bsz, K, T] (t contiguous), W: [N, K] row-major, Y: [Bsz, Hout, T].
// One wave computes a 32(t) x (NT*16)(h) tile of C with V_WMMA_F32_16X16X4_F32,
// stepping K by 4:  2 A-fragments x NT B-fragments -> 2*NT WMMA per k-step.
// EXEC stays all-ones through the WMMA loop (loads are clamped, only stores
// are predicated), as the ISA requires.
//
// Fragment layouts (CDNA5 ISA 7.12.2, 32-bit 16x4 A / 4x16 B / 16x16 C):
//   A: lanes 0-15 = M rows, v0/v1 = K+0/K+1 ; lanes 16-31 same M, K+2/K+3
//   B: lanes 0-15 = N cols, v0/v1 = K+0/K+1 ; lanes 16-31 same N, K+2/K+3
//   C: VGPR r: lanes 0-15 -> (M=r, N=lane), lanes 16-31 -> (M=r+8, N=lane-16)
//
// Addressing: wave-uniform base pointers + per-lane 32-bit element offsets
// (max offset K*T < 2^20) advanced by a uniform stride each iteration, so the
// backend can use saddr + 32-bit voffset global loads instead of 64-bit VALU
// address adds.
// ---------------------------------------------------------------------------
template <int ACT, int NT>
__global__ __launch_bounds__(256) void gemm_bt_wmma(
    const float* __restrict__ A, const float* __restrict__ W,
    const float* __restrict__ bias, float* __restrict__ Y,
    int Bsz, int T, int K, int N, int Hout, int TM, int NB) {
  const int lane = threadIdx.x & 31;
  const int wave = blockIdx.x * (blockDim.x >> 5) + (threadIdx.x >> 5);
  const int nb   = wave % NB;
  const int rest = wave / NB;
  const int tb   = rest % TM;
  const int b    = rest / TM;
  if (b >= Bsz) return;

  const int t0 = tb * 32;
  const int h0 = nb * (NT * 16);
  const int m  = lane & 15;   // M-row (A/C) == N-col (B/C) sub-index
  const int kp = lane >> 4;   // selects K pair {0,1} vs {2,3}

  // Clamped edge indices (rows/cols beyond T/N are computed but never stored).
  int tA0 = t0 + m;      if (tA0 > T - 1) tA0 = T - 1;
  int tA1 = t0 + 16 + m; if (tA1 > T - 1) tA1 = T - 1;
  int hB[NT];
#pragma unroll
  for (int ni = 0; ni < NT; ++ni) {
    int h = h0 + ni * 16 + m;
    hB[ni] = (h > N - 1) ? (N - 1) : h;
  }

  const float* Ab = A + (size_t)b * K * T;  // wave-uniform base
  uint32_t aoff0 = (uint32_t)(2 * kp) * T + tA0;  // element offsets, 32-bit
  uint32_t aoff1 = (uint32_t)(2 * kp) * T + tA1;
  uint32_t woff[NT];
#pragma unroll
  for (int ni = 0; ni < NT; ++ni)
    woff[ni] = (uint32_t)hB[ni] * K + 2 * kp;  // even -> 8B-aligned v2f loads

  v8f acc[2][NT];
#pragma unroll
  for (int mi = 0; mi < 2; ++mi)
#pragma unroll
    for (int ni = 0; ni < NT; ++ni) acc[mi][ni] = (v8f){};

  const uint32_t astep = 4u * T;
  for (int k = 0; k < K; k += 4) {  // K % 4 == 0 for all layers here
    v2f af[2];
    af[0].x = Ab[aoff0];
    af[0].y = Ab[aoff0 + T];
    af[1].x = Ab[aoff1];
    af[1].y = Ab[aoff1 + T];
    v2f bf[NT];
#pragma unroll
    for (int ni = 0; ni < NT; ++ni) bf[ni] = *(const v2f*)(W + woff[ni]);
#pragma unroll
    for (int mi = 0; mi < 2; ++mi)
#pragma unroll
      for (int ni = 0; ni < NT; ++ni)
        acc[mi][ni] = __builtin_amdgcn_wmma_f32_16x16x4_f32(
            false, af[mi], false, bf[ni], (short)0, acc[mi][ni], false, false);
    aoff0 += astep;
    aoff1 += astep;
#pragma unroll
    for (int ni = 0; ni < NT; ++ni) woff[ni] += 4u;
  }

  // Epilogue: bias (+ sigmoid for layer 1), predicated edge stores.
#pragma unroll
  for (int ni = 0; ni < NT; ++ni) {
    const int h = h0 + ni * 16 + m;
    if (h >= N) continue;
    const float bv = bias[h];
    float* yp = Y + ((size_t)b * Hout + h) * T;
#pragma unroll
    for (int mi = 0; mi < 2; ++mi) {
#pragma unroll
      for (int r = 0; r < 8; ++r) {
        const int t = t0 + mi * 16 + r + 8 * kp;
        if (t < T) {
          float v = acc[mi][ni][r] + bv;
          if (ACT) v = 1.0f / (1.0f + expf(-v));
          yp[t] = v;
        }
      }
    }
  }
}

// ---------------------------------------------------------------------------
// dual_exp_iir, in place: x[t] <- a1*p1 + a2*p2 + V0*x[t];  (p2,p1) <- (p1,cur)
// One thread per (b, h) row of T contiguous floats.
// ---------------------------------------------------------------------------
__global__ __launch_bounds__(256) void iir_kernel(
    float* __restrict__ X, const float* __restrict__ A1,
    const float* __restrict__ A2, int rows, int T, int H, float V0f) {
  const int r = blockIdx.x * blockDim.x + threadIdx.x;
  if (r >= rows) return;
  const int h = r % H;
  const float a1 = A1[h], a2 = A2[h];
  float* p = X + (size_t)r * T;
  float p1 = 0.f, p2 = 0.f;
  for (int t = 0; t < T; ++t) {
    const float cur = fmaf(a1, p1, fmaf(a2, p2, V0f * p[t]));
    p[t] = cur;
    p2 = p1;
    p1 = cur;
  }
}

// ---------------------------------------------------------------------------
// LIF scan: v <- decay*v + psp[t]; s = (v > 1); store s; v -= s  (soft reset)
// PSP rows may live in an Hpad-padded buffer; output rows are dense (Hreal).
// Safe for in-place use when Hreal == Hpad and PSP == S.
// ---------------------------------------------------------------------------
__global__ __launch_bounds__(256) void lif_kernel(
    const float* __restrict__ PSP, float* __restrict__ S, int rows, int T,
    int Hreal, int Hpad, float decay) {
  const int r = blockIdx.x * blockDim.x + threadIdx.x;
  if (r >= rows) return;
  const int b = r / Hreal;
  const int o = r % Hreal;
  const float* p = PSP + ((size_t)b * Hpad + o) * T;
  float* s = S + (size_t)r * T;
  float v = 0.f;
  for (int t = 0; t < T; ++t) {
    v = fmaf(decay, v, p[t]);
    const float sp = (v > 1.0f) ? 1.0f : 0.0f;
    s[t] = sp;
    v -= sp;
  }
}

extern "C" void kernel_launch(void* const* d_in, const int* in_sizes, int n_in,
                              void* d_out, int out_size, void* d_ws,
                              size_t ws_size, hipStream_t stream) {
  const float* inputs = (const float*)d_in[0];
  const float* w1     = (const float*)d_in[1];
  const float* b1     = (const float*)d_in[2];
  const float* a1_2   = (const float*)d_in[3];
  const float* a2_2   = (const float*)d_in[4];
  const float* w2     = (const float*)d_in[5];
  const float* b2     = (const float*)d_in[6];
  const float* a1_3   = (const float*)d_in[7];
  const float* a2_3   = (const float*)d_in[8];
  const float* w3     = (const float*)d_in[9];
  const float* b3     = (const float*)d_in[10];
  float* out = (float*)d_out;

  const int B = 256, IN = 784, H = 500, OUT = 10, T = 100, HP = 16;

  float* buf1 = (float*)d_ws;              // [B,H,T] 51.2 MB (also psp3 pad)
  float* buf2 = buf1 + (size_t)B * H * T;  // [B,H,T] 51.2 MB

  const double TAU_M = 4.0, TAU_S = 1.0;
  const double ETA = TAU_M / TAU_S;
  const float V0f = (float)(pow(ETA, ETA / (ETA - 1.0)) / (ETA - 1.0));
  const float DM  = (float)exp(-1.0 / TAU_M);

  const int TM = (T + 31) / 32;  // 4 time tiles of 32

  // 1) ann = sigmoid(inputs . w1^T + b1)  -> buf1   (M=25600,K=784,N=500)
  {
    const int NB = (H + 63) / 64;  // 64-wide n tiles (NT=4)
    const int waves = B * TM * NB;
    gemm_bt_wmma<1, 4><<<(waves + 7) / 8, 256, 0, stream>>>(
        inputs, w1, b1, buf1, B, T, IN, H, H, TM, NB);
  }
  // 2) axon2: dual-exp IIR in place on buf1
  {
    const int rows = B * H;
    iir_kernel<<<(rows + 255) / 256, 256, 0, stream>>>(buf1, a1_2, a2_2, rows,
                                                       T, H, V0f);
  }
  // 3) psp2 = x . w2^T + b2 -> buf2   (M=25600,K=500,N=500)
  {
    const int NB = (H + 63) / 64;
    const int waves = B * TM * NB;
    gemm_bt_wmma<0, 4><<<(waves + 7) / 8, 256, 0, stream>>>(
        buf1, w2, b2, buf2, B, T, H, H, H, TM, NB);
  }
  // 4) snn2 LIF in place on buf2
  {
    const int rows = B * H;
    lif_kernel<<<(rows + 255) / 256, 256, 0, stream>>>(buf2, buf2, rows, T, H,
                                                       H, DM);
  }
  // 5) axon3: IIR in place on buf2
  {
    const int rows = B * H;
    iir_kernel<<<(rows + 255) / 256, 256, 0, stream>>>(buf2, a1_3, a2_3, rows,
                                                       T, H, V0f);
  }
  // 6) psp3 = x . w3^T + b3 -> buf1 (padded to HP=16 rows per batch; NT=1)
  {
    const int NB = 1;  // N=10 fits one 16-wide tile
    const int waves = B * TM * NB;
    gemm_bt_wmma<0, 1><<<(waves + 7) / 8, 256, 0, stream>>>(
        buf2, w3, b3, buf1, B, T, H, OUT, HP, TM, NB);
  }
  // 7) snn3 LIF -> d_out [B, 10, T]
  {
    const int rows = B * OUT;
    lif_kernel<<<(rows + 255) / 256, 256, 0, stream>>>(buf1, out, rows, T, OUT,
                                                       HP, DM);
  }
}